// AECR_model_78529182040603
// MI455X (gfx1250) — compile-verified
//
#include <hip/hip_runtime.h>
#include <math.h>

#define CDIM   256
#define HWDIM  4096
#define KNOWN  3072
#define MPTS   1024
#define BATCH  4
#define EPSV   1e-8f

#define BM 128
#define BN 128
#define KC 32

typedef float v2f __attribute__((ext_vector_type(2)));
typedef float v8f __attribute__((ext_vector_type(8)));

__device__ __forceinline__ unsigned f2ord(float f) {
    unsigned u = __float_as_uint(f);
    return (u & 0x80000000u) ? ~u : (u | 0x80000000u);
}
__device__ __forceinline__ float ord2f(unsigned u) {
    unsigned v = (u & 0x80000000u) ? (u ^ 0x80000000u) : ~u;
    return __uint_as_float(v);
}

// ---------------------------------------------------------------- zero init
__global__ void zero_kernel(unsigned long long* p, int n) {
    int i = blockIdx.x * blockDim.x + threadIdx.x;
    if (i < n) p[i] = 0ull;
}

// ------------------------------------------------- per-known-column 1/(||.||+eps)
__global__ void norm_kernel(const float* __restrict__ x,
                            const int* __restrict__ non_idx,
                            float* __restrict__ inv_norm) {
    int b = blockIdx.y;
    int k = blockIdx.x * blockDim.x + threadIdx.x;
    if (k >= KNOWN) return;
    const float* X = x + (size_t)b * CDIM * HWDIM;
    int p = non_idx[k];
    float s = 0.f;
    for (int c = 0; c < CDIM; ++c) {
        float v = X[(size_t)c * HWDIM + p];
        s += v * v;
    }
    inv_norm[b * KNOWN + k] = 1.0f / (sqrtf(s) + EPSV);
}

// --------------------------- fused sim-GEMM + row argmax (fp32 WMMA 16x16x4)
__global__ __launch_bounds__(256)
void simmax_kernel(const float* __restrict__ x,
                   const int* __restrict__ non_idx,
                   const float* __restrict__ inv_norm,
                   unsigned long long* __restrict__ vm_packed) {
    __shared__ float A_lds[KC * BM];   // (c-pair)-interleaved: [(c>>1)][row][c&1]
    __shared__ float B_lds[KC * BN];
    __shared__ int   colp[BN];

    const int b        = blockIdx.z;
    const int rowblock = blockIdx.x * BM;
    const int colblock = blockIdx.y * BN;
    const float* X = x + (size_t)b * CDIM * HWDIM;

    const int tid  = threadIdx.x;
    const int lane = tid & 31;
    const int wave = tid >> 5;
    const int wrow = wave & 3;   // 4 row groups of 32 rows (2 row-tiles each)
    const int wcol = wave >> 2;  // 2 col groups of 64 cols (4 col-tiles each)

    if (tid < BN) colp[tid] = non_idx[colblock + tid];
    __syncthreads();

    v8f acc[2][4] = {};

    const int m       = lane & 15;
    const int chi_off = (lane >> 4) << 1;   // 0 or 2 (K sub-row of WMMA frag)

    for (int cb = 0; cb < CDIM; cb += KC) {
        __syncthreads();
        // cooperatively stage A chunk: KC x BM (4096 floats, 16 per thread)
        #pragma unroll
        for (int i = 0; i < (KC * BM) / 256; ++i) {
            int linear = tid + i * 256;
            int row = linear & (BM - 1);
            int c   = linear >> 7;
            float v = X[(size_t)(cb + c) * HWDIM + rowblock + row];
            A_lds[(c >> 1) * (2 * BM) + row * 2 + (c & 1)] = v;
        }
        // stage B chunk: KC x BN (gathered known columns)
        #pragma unroll
        for (int i = 0; i < (KC * BN) / 256; ++i) {
            int linear = tid + i * 256;
            int col = linear & (BN - 1);
            int c   = linear >> 7;
            float v = X[(size_t)(cb + c) * HWDIM + colp[col]];
            B_lds[(c >> 1) * (2 * BN) + col * 2 + (c & 1)] = v;
        }
        __syncthreads();

        #pragma unroll
        for (int kk = 0; kk < KC; kk += 4) {
            int chi = kk + chi_off;          // even
            v2f afrag[2], bfrag[4];
            #pragma unroll
            for (int rt = 0; rt < 2; ++rt) {
                int row = wrow * 32 + rt * 16 + m;
                afrag[rt] = *(const v2f*)&A_lds[(chi >> 1) * (2 * BM) + row * 2];
            }
            #pragma unroll
            for (int ct = 0; ct < 4; ++ct) {
                int col = wcol * 64 + ct * 16 + m;
                bfrag[ct] = *(const v2f*)&B_lds[(chi >> 1) * (2 * BN) + col * 2];
            }
            #pragma unroll
            for (int rt = 0; rt < 2; ++rt)
                #pragma unroll
                for (int ct = 0; ct < 4; ++ct)
                    acc[rt][ct] = __builtin_amdgcn_wmma_f32_16x16x4_f32(
                        false, afrag[rt], false, bfrag[ct],
                        (short)0, acc[rt][ct], false, false);
        }
    }

    // scale by column inv-norm, per-row argmax, global combine
    const int hi = lane >> 4;
    #pragma unroll
    for (int rt = 0; rt < 2; ++rt) {
        float bestv[8];
        int   bestc[8];
        #pragma unroll
        for (int r = 0; r < 8; ++r) { bestv[r] = -INFINITY; bestc[r] = 0x7FFFFFFF; }
        #pragma unroll
        for (int ct = 0; ct < 4; ++ct) {
            int col = colblock + wcol * 64 + ct * 16 + m;   // global known-col index
            float s = inv_norm[b * KNOWN + col];
            #pragma unroll
            for (int r = 0; r < 8; ++r) {
                float v = acc[rt][ct][r] * s;
                if (v > bestv[r] || (v == bestv[r] && col < bestc[r])) {
                    bestv[r] = v; bestc[r] = col;
                }
            }
        }
        #pragma unroll
        for (int r = 0; r < 8; ++r) {
            float v = bestv[r];
            int   c = bestc[r];
            // reduce across the 16 lanes that hold this row's 16 columns
            #pragma unroll
            for (int off = 8; off >= 1; off >>= 1) {
                float ov = __shfl_xor(v, off, 16);
                int   oc = __shfl_xor(c, off, 16);
                if (ov > v || (ov == v && oc < c)) { v = ov; c = oc; }
            }
            if (m == 0) {
                int row = rowblock + wrow * 32 + rt * 16 + r + 8 * hi;
                unsigned long long key =
                    ((unsigned long long)f2ord(v) << 32) | (unsigned)(~(unsigned)c);
                atomicMax(&vm_packed[(size_t)b * HWDIM + row], key);
            }
        }
    }
}

// ---------------------------------------------------------------- unpack
__global__ void unpack_kernel(const unsigned long long* __restrict__ vm_packed,
                              float* __restrict__ vmax, int* __restrict__ ind) {
    int i = blockIdx.x * blockDim.x + threadIdx.x;
    if (i >= BATCH * HWDIM) return;
    unsigned long long key = vm_packed[i];
    vmax[i] = ord2f((unsigned)(key >> 32));
    ind[i]  = (int)(~(unsigned)(key & 0xFFFFFFFFull));
}

// -------------------------------------- known rows: out[p] = xf[non_idx[ind[p]]]
__global__ void gather_kernel(const float* __restrict__ x,
                              const int* __restrict__ non_idx,
                              const int* __restrict__ ind,
                              float* __restrict__ out) {
    int b = blockIdx.y;
    int t = blockIdx.x * blockDim.x + threadIdx.x;
    if (t >= KNOWN) return;
    int p   = non_idx[t];
    int src = non_idx[ind[b * HWDIM + p]];
    const float* X = x   + (size_t)b * CDIM * HWDIM;
    float*       O = out + (size_t)b * CDIM * HWDIM;
    for (int c = 0; c < CDIM; ++c)
        O[(size_t)c * HWDIM + p] = X[(size_t)c * HWDIM + src];
}

// -------------------- sequential recursive blend: one wave per batch
__global__ void scan_kernel(const float* __restrict__ x,
                            const int* __restrict__ mask_idx,
                            const int* __restrict__ non_idx,
                            const float* __restrict__ vmax,
                            const int* __restrict__ ind,
                            float* __restrict__ out) {
    int b    = blockIdx.x;
    int lane = threadIdx.x;          // 32 lanes, 8 channels each = C=256
    const float* X = x   + (size_t)b * CDIM * HWDIM;
    float*       O = out + (size_t)b * CDIM * HWDIM;

    float prev[8];
    #pragma unroll
    for (int j = 0; j < 8; ++j) prev[j] = 0.f;

    for (int i = 0; i < MPTS; ++i) {
        int   mi  = mask_idx[i];
        float vm  = vmax[b * HWDIM + mi];
        int   cch = non_idx[ind[b * HWDIM + mi]];

        float u[8];
        float n2 = 0.f, dp = 0.f;
        #pragma unroll
        for (int j = 0; j < 8; ++j) {
            u[j] = X[(size_t)(j * 32 + lane) * HWDIM + mi];
            n2 += u[j] * u[j];
            dp += prev[j] * u[j];
        }
        #pragma unroll
        for (int off = 16; off >= 1; off >>= 1) {
            n2 += __shfl_xor(n2, off, 32);
            dp += __shfl_xor(dp, off, 32);
        }
        float at = dp / (sqrtf(n2) + EPSV);
        float a_new, a_ori;
        if (i == 0) { a_new = 0.f; a_ori = 1.f; }
        else        { float denom = at + vm; a_new = at / denom; a_ori = vm / denom; }

        #pragma unroll
        for (int j = 0; j < 8; ++j) {
            float km = X[(size_t)(j * 32 + lane) * HWDIM + cch];
            prev[j] = a_new * prev[j] + a_ori * km;
            O[(size_t)(j * 32 + lane) * HWDIM + mi] = prev[j];
        }
    }
}

extern "C" void kernel_launch(void* const* d_in, const int* in_sizes, int n_in,
                              void* d_out, int out_size, void* d_ws, size_t ws_size,
                              hipStream_t stream) {
    const float* x        = (const float*)d_in[0];
    const int*   mask_idx = (const int*)d_in[1];
    const int*   non_idx  = (const int*)d_in[2];
    float*       out      = (float*)d_out;

    char* ws = (char*)d_ws;
    unsigned long long* vm_packed = (unsigned long long*)ws;          // 4*4096*8  = 131072 B
    float* inv_norm = (float*)(ws + 131072);                          // 4*3072*4  =  49152 B
    float* vmax     = (float*)(ws + 131072 + 49152);                  // 4*4096*4  =  65536 B
    int*   ind      = (int*)  (ws + 131072 + 49152 + 65536);          // 4*4096*4  =  65536 B

    zero_kernel<<<(BATCH * HWDIM + 255) / 256, 256, 0, stream>>>(vm_packed, BATCH * HWDIM);

    {
        dim3 g((KNOWN + 255) / 256, BATCH);
        norm_kernel<<<g, 256, 0, stream>>>(x, non_idx, inv_norm);
    }
    {
        dim3 g(HWDIM / BM, KNOWN / BN, BATCH);   // 32 x 24 x 4
        simmax_kernel<<<g, 256, 0, stream>>>(x, non_idx, inv_norm, vm_packed);
    }
    unpack_kernel<<<(BATCH * HWDIM + 255) / 256, 256, 0, stream>>>(vm_packed, vmax, ind);
    {
        dim3 g((KNOWN + 255) / 256, BATCH);
        gather_kernel<<<g, 256, 0, stream>>>(x, non_idx, ind, out);
    }
    scan_kernel<<<BATCH, 32, 0, stream>>>(x, mask_idx, non_idx, vmax, ind, out);
}